// Proposals_82231443849475
// MI455X (gfx1250) — compile-verified
//
#include <hip/hip_runtime.h>

#ifndef __has_builtin
#define __has_builtin(x) 0
#endif

#if __has_builtin(__builtin_amdgcn_global_load_async_to_lds_b128)
#define HAVE_ASYNC_LDS 1
#else
#define HAVE_ASYNC_LDS 0
#endif
#if __has_builtin(__builtin_amdgcn_s_wait_asynccnt)
#define HAVE_WAIT_ASYNC 1
#else
#define HAVE_WAIT_ASYNC 0
#endif

#define BATCH 8
#define NANCH 262144
#define KTOP  6000
#define PROP  1000
#define NB1   4096
#define NB3   256
#define NEGF  (-1.0e9f)
#define NMSTHR 0.7f
#define NMS_TPB 1024
#define ITEMS 6               // 6*1024 = 6144 >= 6000

#if HAVE_ASYNC_LDS
// Matches the builtin's parameter pointee type exactly:
// 'int __attribute__((__vector_size__(4 * sizeof(int))))'
typedef int async_v4i __attribute__((vector_size(16)));
typedef __attribute__((address_space(1))) async_v4i* async_gp_t;
typedef __attribute__((address_space(3))) async_v4i* async_lp_t;
#endif

// monotonic float -> uint transform (order-preserving for all floats)
__device__ __forceinline__ unsigned ordbits(float f) {
    unsigned b = __float_as_uint(f);
    return b ^ ((b & 0x80000000u) ? 0xFFFFFFFFu : 0x80000000u);
}

// ---------------------------------------------------------------- init
__global__ void k_init(unsigned* h1, unsigned* h2, unsigned* h3,
                       unsigned* cnts, float* selS, float* selB) {
    int i0 = blockIdx.x * blockDim.x + threadIdx.x;
    int stride = gridDim.x * blockDim.x;
    for (int i = i0; i < BATCH * NB1; i += stride) { h1[i] = 0u; h2[i] = 0u; }
    for (int i = i0; i < BATCH * NB3; i += stride) h3[i] = 0u;
    for (int i = i0; i < BATCH * 2;   i += stride) cnts[i] = 0u;
    for (int i = i0; i < BATCH * KTOP; i += stride) selS[i] = NEGF;
    for (int i = i0; i < BATCH * KTOP * 4; i += stride) selB[i] = 0.0f;
}

// ---------------------------------------------------------------- pass 1: hist on top 12 bits (LDS-privatized)
#define H1_ELEMS 32
#define H1_CHUNK (H1_ELEMS * 256)   // 8192 elements per block
__global__ void k_hist1(const float* __restrict__ probs, unsigned* __restrict__ hist) {
    __shared__ unsigned lh[NB1];
    for (int i = threadIdx.x; i < NB1; i += blockDim.x) lh[i] = 0u;
    __syncthreads();
    const int b = blockIdx.y;
    const size_t base = (size_t)b * NANCH + (size_t)blockIdx.x * H1_CHUNK;
    for (int e = 0; e < H1_ELEMS; ++e) {
        size_t idx = base + e * 256 + threadIdx.x;
        __builtin_prefetch(probs + (idx + 2048) * 2 + 1, 0, 1); // global_prefetch_b8
        float s = probs[idx * 2 + 1];
        atomicAdd(&lh[ordbits(s) >> 20], 1u);                   // ds_add
    }
    __syncthreads();
    for (int i = threadIdx.x; i < NB1; i += blockDim.x)
        if (lh[i]) atomicAdd(&hist[b * NB1 + i], lh[i]);
}

__global__ void k_scan1(const unsigned* __restrict__ h, int* __restrict__ tinfo) {
    if (threadIdx.x != 0) return;
    const int b = blockIdx.x;
    unsigned cum = 0; int thr = 0; int quota = KTOP;
    for (int i = NB1 - 1; i >= 0; --i) {
        unsigned c = h[b * NB1 + i];
        if (cum + c >= (unsigned)KTOP) { thr = i; quota = KTOP - (int)cum; break; }
        cum += c;
    }
    tinfo[b * 8 + 0] = thr; tinfo[b * 8 + 1] = quota;
}

// ---------------------------------------------------------------- pass 2: refine middle 12 bits
__global__ void k_hist2(const float* __restrict__ probs, const int* __restrict__ tinfo,
                        unsigned* __restrict__ h2) {
    const int b = blockIdx.y;
    const size_t a = (size_t)blockIdx.x * blockDim.x + threadIdx.x;
    const unsigned thr = (unsigned)tinfo[b * 8 + 0];
    unsigned u = ordbits(probs[((size_t)b * NANCH + a) * 2 + 1]);
    if ((u >> 20) == thr) atomicAdd(&h2[b * NB1 + ((u >> 8) & 0xFFFu)], 1u);
}

__global__ void k_scan2(const unsigned* __restrict__ h2, int* __restrict__ tinfo) {
    if (threadIdx.x != 0) return;
    const int b = blockIdx.x;
    const int thr12 = tinfo[b * 8 + 0];
    const int quota1 = tinfo[b * 8 + 1];
    unsigned cum = 0; int mid = 0; int q2 = quota1;
    for (int i = NB1 - 1; i >= 0; --i) {
        unsigned c = h2[b * NB1 + i];
        if ((int)(cum + c) >= quota1) { mid = i; q2 = quota1 - (int)cum; break; }
        cum += c;
    }
    tinfo[b * 8 + 2] = (thr12 << 12) | mid; tinfo[b * 8 + 3] = q2;
}

// ---------------------------------------------------------------- pass 3: refine low 8 bits -> exact threshold
__global__ void k_hist3(const float* __restrict__ probs, const int* __restrict__ tinfo,
                        unsigned* __restrict__ h3) {
    const int b = blockIdx.y;
    const size_t a = (size_t)blockIdx.x * blockDim.x + threadIdx.x;
    const unsigned thr24 = (unsigned)tinfo[b * 8 + 2];
    unsigned u = ordbits(probs[((size_t)b * NANCH + a) * 2 + 1]);
    if ((u >> 8) == thr24) atomicAdd(&h3[b * NB3 + (u & 0xFFu)], 1u);
}

__global__ void k_scan3(const unsigned* __restrict__ h3, int* __restrict__ tinfo) {
    if (threadIdx.x != 0) return;
    const int b = blockIdx.x;
    const unsigned thr24 = (unsigned)tinfo[b * 8 + 2];
    const int quota2 = tinfo[b * 8 + 3];
    unsigned cum = 0; int low = 0; int q3 = quota2;
    for (int i = NB3 - 1; i >= 0; --i) {
        unsigned c = h3[b * NB3 + i];
        if ((int)(cum + c) >= quota2) { low = i; q3 = quota2 - (int)cum; break; }
        cum += c;
    }
    tinfo[b * 8 + 4] = (int)((thr24 << 8) | (unsigned)low);
    tinfo[b * 8 + 5] = q3;
}

// ---------------------------------------------------------------- compaction + box decode (reference quirk: d = anchor)
__global__ void k_compact(const float* __restrict__ probs, const float* __restrict__ anchors,
                          const int* __restrict__ tinfo, unsigned* __restrict__ cnts,
                          float* __restrict__ selS, float* __restrict__ selB) {
    const int b = blockIdx.y;
    const size_t a = (size_t)blockIdx.x * blockDim.x + threadIdx.x;
    const float s = probs[((size_t)b * NANCH + a) * 2 + 1];
    const unsigned u = ordbits(s);
    const unsigned ustar = (unsigned)tinfo[b * 8 + 4];
    const int q3 = tinfo[b * 8 + 5];
    int dest = -1;
    if (u > ustar) {
        dest = (int)atomicAdd(&cnts[b * 2 + 0], 1u);
    } else if (u == ustar) {
        unsigned p = atomicAdd(&cnts[b * 2 + 1], 1u);
        if ((int)p < q3) dest = KTOP - 1 - (int)p;
    }
    if (dest < 0 || dest >= KTOP) return;

    const float* an = anchors + ((size_t)b * NANCH + a) * 4;
    float y1a = an[0], x1a = an[1], y2a = an[2], x2a = an[3];
    float h = y2a - y1a, w = x2a - x1a;
    float cy = y1a + 0.5f * h + y1a * h;      // d == anchor in the reference
    float cx = x1a + 0.5f * w + x1a * w;
    float h2 = h * expf(y2a);
    float w2 = w * expf(x2a);
    float y1 = cy - 0.5f * h2, x1 = cx - 0.5f * w2;
    float y2 = y1 + h2,        x2 = x1 + w2;
    y1 = fminf(fmaxf(y1, 0.0f), 1.0f); x1 = fminf(fmaxf(x1, 0.0f), 1.0f);
    y2 = fminf(fmaxf(y2, 0.0f), 1.0f); x2 = fminf(fmaxf(x2, 0.0f), 1.0f);

    const size_t o = (size_t)b * KTOP + dest;
    selS[o] = s;
    float4 bb; bb.x = y1; bb.y = x1; bb.z = y2; bb.w = x2;
    ((float4*)selB)[o] = bb;
}

// ---------------------------------------------------------------- NMS: 1 workgroup per batch, 32 waves
__global__ void __launch_bounds__(NMS_TPB)
k_nms(const float* __restrict__ selS, const float* __restrict__ selB, float* __restrict__ out) {
    __shared__ __align__(16) float s_sc[ITEMS * NMS_TPB];
    __shared__ float red_s[32];
    __shared__ int   red_i[32];
    __shared__ float s_wbox[4];
    __shared__ int   s_widx;
    __shared__ float s_wsc;

    const int b = blockIdx.x;
    const int tid = threadIdx.x;

    // ---- stage scores global -> LDS through the async tensor/DMA path
    const char* gbase = (const char*)(selS + (size_t)b * KTOP);
    char* lbase = (char*)s_sc;
#if HAVE_ASYNC_LDS
    for (int c = tid; c < (KTOP * 4) / 16; c += NMS_TPB) {
        __builtin_amdgcn_global_load_async_to_lds_b128(
            (async_gp_t)(gbase + 16 * c),
            (async_lp_t)(lbase + 16 * c), 0, 0);
    }
#if HAVE_WAIT_ASYNC
    __builtin_amdgcn_s_wait_asynccnt(0);
#else
    asm volatile("s_wait_asynccnt 0" ::: "memory");
#endif
#else
    for (int c = tid; c < (KTOP * 4) / 16; c += NMS_TPB)
        ((float4*)lbase)[c] = ((const float4*)gbase)[c];
#endif
    __syncthreads();

    // ---- registers: 6 (score, box) entries per thread, strided layout j = i*1024 + tid
    float sc[ITEMS], by1[ITEMS], bx1[ITEMS], by2[ITEMS], bx2[ITEMS];
    for (int i = 0; i < ITEMS; ++i) {
        int j = i * NMS_TPB + tid;
        if (j < KTOP) {
            sc[i] = s_sc[j];
            float4 bb = ((const float4*)selB)[(size_t)b * KTOP + j];
            by1[i] = bb.x; bx1[i] = bb.y; by2[i] = bb.z; bx2[i] = bb.w;
        } else {
            sc[i] = NEGF; by1[i] = bx1[i] = by2[i] = bx2[i] = 0.0f;
        }
    }
    __syncthreads();

    const int lane = tid & 31, wid = tid >> 5;
    for (int it = 0; it < PROP; ++it) {
        // local argmax
        float best = sc[0]; int bidx = tid;
        for (int i = 1; i < ITEMS; ++i) {
            int j = i * NMS_TPB + tid;
            if (sc[i] > best) { best = sc[i]; bidx = j; }
        }
        // wave32 shuffle reduction
        for (int off = 16; off > 0; off >>= 1) {
            float os = __shfl_down(best, off, 32);
            int   oi = __shfl_down(bidx, off, 32);
            if (os > best || (os == best && oi < bidx)) { best = os; bidx = oi; }
        }
        if (lane == 0) { red_s[wid] = best; red_i[wid] = bidx; }
        __syncthreads();
        if (tid < 32) {
            best = red_s[tid]; bidx = red_i[tid];
            for (int off = 16; off > 0; off >>= 1) {
                float os = __shfl_down(best, off, 32);
                int   oi = __shfl_down(bidx, off, 32);
                if (os > best || (os == best && oi < bidx)) { best = os; bidx = oi; }
            }
            if (tid == 0) { s_widx = bidx; s_wsc = best; }
        }
        __syncthreads();

        const int widx = s_widx;
        const float wsc = s_wsc;
        const bool valid = wsc > NEGF * 0.5f;
        if (tid == (widx & (NMS_TPB - 1))) {       // owner publishes winner box
            int i = widx >> 10;
            s_wbox[0] = by1[i]; s_wbox[1] = bx1[i]; s_wbox[2] = by2[i]; s_wbox[3] = bx2[i];
            sc[i] = NEGF;                           // winner self-suppressed
        }
        __syncthreads();

        if (tid == 0) {
            float4 o;
            if (valid) { o.x = s_wbox[0]; o.y = s_wbox[1]; o.z = s_wbox[2]; o.w = s_wbox[3]; }
            else       { o.x = o.y = o.z = o.w = 0.0f; }
            ((float4*)out)[(size_t)b * PROP + it] = o;
        }
        if (valid) {
            const float wy1 = s_wbox[0], wx1 = s_wbox[1], wy2 = s_wbox[2], wx2 = s_wbox[3];
            const float warea = (wy2 - wy1) * (wx2 - wx1);
            for (int i = 0; i < ITEMS; ++i) {
                float iy1 = fmaxf(wy1, by1[i]);
                float ix1 = fmaxf(wx1, bx1[i]);
                float iy2 = fminf(wy2, by2[i]);
                float ix2 = fminf(wx2, bx2[i]);
                float inter = fmaxf(iy2 - iy1, 0.0f) * fmaxf(ix2 - ix1, 0.0f);
                float ai = (by2[i] - by1[i]) * (bx2[i] - bx1[i]);
                float iou = inter / (ai + warea - inter + 1e-8f);
                if (iou > NMSTHR) sc[i] = NEGF;
            }
        }
        __syncthreads();
    }
}

// ---------------------------------------------------------------- launch
extern "C" void kernel_launch(void* const* d_in, const int* in_sizes, int n_in,
                              void* d_out, int out_size, void* d_ws, size_t ws_size,
                              hipStream_t stream) {
    const float* probs   = (const float*)d_in[0];  // (8, 262144, 2)
    const float* rpnbbox = (const float*)d_in[1];  // dead in reference (d = ga quirk)
    const float* anchors = (const float*)d_in[2];  // (8, 262144, 4)
    (void)rpnbbox; (void)in_sizes; (void)n_in; (void)out_size; (void)ws_size;
    float* out = (float*)d_out;                    // (8, 1000, 4)

    unsigned char* w = (unsigned char*)d_ws;
    unsigned* h1    = (unsigned*)w; w += (size_t)BATCH * NB1 * 4;
    unsigned* h2    = (unsigned*)w; w += (size_t)BATCH * NB1 * 4;
    unsigned* h3    = (unsigned*)w; w += (size_t)BATCH * NB3 * 4;
    int*      tinfo = (int*)w;      w += (size_t)BATCH * 8 * 4;
    unsigned* cnts  = (unsigned*)w; w += (size_t)BATCH * 2 * 4;
    float*    selS  = (float*)w;    w += (size_t)BATCH * KTOP * 4;
    float*    selB  = (float*)w;    w += (size_t)BATCH * KTOP * 16;

    k_init   <<<256, 256, 0, stream>>>(h1, h2, h3, cnts, selS, selB);
    k_hist1  <<<dim3(NANCH / H1_CHUNK, BATCH), 256, 0, stream>>>(probs, h1);
    k_scan1  <<<BATCH, 32, 0, stream>>>(h1, tinfo);
    k_hist2  <<<dim3(NANCH / 256, BATCH), 256, 0, stream>>>(probs, tinfo, h2);
    k_scan2  <<<BATCH, 32, 0, stream>>>(h2, tinfo);
    k_hist3  <<<dim3(NANCH / 256, BATCH), 256, 0, stream>>>(probs, tinfo, h3);
    k_scan3  <<<BATCH, 32, 0, stream>>>(h3, tinfo);
    k_compact<<<dim3(NANCH / 256, BATCH), 256, 0, stream>>>(probs, anchors, tinfo, cnts, selS, selB);
    k_nms    <<<BATCH, NMS_TPB, 0, stream>>>(selS, selB, out);
}